// GraphBasedPNATModel_46815143527017
// MI455X (gfx1250) — compile-verified
//
#include <hip/hip_runtime.h>
#include <math.h>

// Problem dimensions (fixed by the reference setup)
#define Nn   50000
#define Ee   500000
#define DIN  256
#define Hh   128
#define Ll   3
#define Gg   32

typedef __attribute__((ext_vector_type(2))) float v2f;
typedef __attribute__((ext_vector_type(8))) float v8f;

// ---------------- float atomics (hardware f32 atomic add/max/min) ----------
__device__ __forceinline__ void atomAddF(float* p, float v) {
  __hip_atomic_fetch_add(p, v, __ATOMIC_RELAXED, __HIP_MEMORY_SCOPE_AGENT);
}
__device__ __forceinline__ void atomMaxF(float* p, float v) {
  __hip_atomic_fetch_max(p, v, __ATOMIC_RELAXED, __HIP_MEMORY_SCOPE_AGENT);
}
__device__ __forceinline__ void atomMinF(float* p, float v) {
  __hip_atomic_fetch_min(p, v, __ATOMIC_RELAXED, __HIP_MEMORY_SCOPE_AGENT);
}

// ---------------- generic fp32 WMMA GEMM: C[M,128] = A[M,K] @ B[K,128] -----
// Block: 256 threads (8 waves). Block tile 128x128, K staged in chunks of 16.
// Uses V_WMMA_F32_16X16X4_F32 (wave32). Requires K % 16 == 0, N == 128.
// Both A and B tiles are stored with the fragment-contiguous dimension (K)
// innermost and padded to 20 floats so every fragment is a single aligned
// 8-byte LDS load (ds_load_b64) -- no register repacking before v_wmma.
__global__ __launch_bounds__(256) void gemm_wmma_f32(
    const float* __restrict__ A, int lda,
    const float* __restrict__ B, int ldb,
    float* __restrict__ C, int ldc,
    int M, int K, const float* __restrict__ bias, int accumulate)
{
  __shared__ float As[128][20];   // [row][k]  pad 16->20: float4-aligned, conflict-free
  __shared__ float Bs[128][20];   // [col][k]  transposed at staging time

  const int tid  = threadIdx.x;
  const int wave = tid >> 5;
  const int lane = tid & 31;
  const int lh   = lane >> 4;     // lane half (K split for A/B fragments)
  const int lm   = lane & 15;
  const int m0   = blockIdx.x * 128;

  v8f acc[8];
  #pragma unroll
  for (int i = 0; i < 8; ++i)
    #pragma unroll
    for (int j = 0; j < 8; ++j) acc[i][j] = 0.0f;

  const int ktiles = K >> 4;
  for (int kt = 0; kt < ktiles; ++kt) {
    const int kk = kt << 4;
    // stage A tile: 128 rows x 16 K (zero-fill rows >= M)
    {
      int r = tid >> 1;
      int c = (tid & 1) << 3;
      int row = m0 + r;
      float4 v0, v1;
      if (row < M) {
        const float4* p = reinterpret_cast<const float4*>(A + (size_t)row * lda + kk + c);
        v0 = p[0]; v1 = p[1];
      } else {
        v0 = make_float4(0.f, 0.f, 0.f, 0.f); v1 = v0;
      }
      *reinterpret_cast<float4*>(&As[r][c])     = v0;
      *reinterpret_cast<float4*>(&As[r][c + 4]) = v1;
    }
    // stage B tile: 16 K x 128 cols, transposed into Bs[col][k]
    {
      #pragma unroll
      for (int i = 0; i < 2; ++i) {
        int idx = tid + (i << 8);
        int row = idx >> 5;         // k within tile: 0..15
        int col = (idx & 31) << 2;  // 0..124
        const float4* p = reinterpret_cast<const float4*>(B + (size_t)(kk + row) * ldb + col);
        float4 v = *p;
        Bs[col + 0][row] = v.x;
        Bs[col + 1][row] = v.y;
        Bs[col + 2][row] = v.z;
        Bs[col + 3][row] = v.w;
      }
    }
    __syncthreads();
    #pragma unroll
    for (int k = 0; k < 16; k += 4) {
      // A fragment: lanes 0-15 hold K={k,k+1}, lanes 16-31 K={k+2,k+3}, M = lane&15
      v2f a = *reinterpret_cast<const v2f*>(&As[(wave << 4) + lm][k + (lh << 1)]);
      #pragma unroll
      for (int ct = 0; ct < 8; ++ct) {
        // B fragment: N = lane&15 (+16*ct), K split across lane halves
        v2f b = *reinterpret_cast<const v2f*>(&Bs[(ct << 4) + lm][k + (lh << 1)]);
        acc[ct] = __builtin_amdgcn_wmma_f32_16x16x4_f32(
            false, a, false, b, (short)0, acc[ct], false, false);
      }
    }
    __syncthreads();
  }

  // store: VGPR r -> M = r (lanes 0-15) / r+8 (lanes 16-31), N = lane&15
  #pragma unroll
  for (int ct = 0; ct < 8; ++ct) {
    int col = (ct << 4) + lm;
    float bv = bias ? bias[col] : 0.0f;
    #pragma unroll
    for (int r = 0; r < 8; ++r) {
      int row = m0 + (wave << 4) + r + (lh << 3);
      if (row < M) {
        size_t off = (size_t)row * ldc + col;
        float v = acc[ct][r] + bv;
        if (accumulate) C[off] += v; else C[off] = v;
      }
    }
  }
}

// ---------------- small utility kernels ------------------------------------
__global__ void fill_kernel(float* p, long long n, float v) {
  long long i = (long long)blockIdx.x * 256 + threadIdx.x;
  if (i < n) p[i] = v;
}

__global__ void deg_kernel(const int* __restrict__ dst, float* __restrict__ deg, int E) {
  int e = blockIdx.x * 256 + threadIdx.x;
  if (e < E) atomAddF(&deg[dst[e]], 1.0f);
}

__global__ void avglog_kernel(const float* __restrict__ deg, float* avgp) {
  int i = blockIdx.x * blockDim.x + threadIdx.x;
  int stride = gridDim.x * blockDim.x;
  float s = 0.f;
  for (int n = i; n < Nn; n += stride) s += log1pf(deg[n]);
  atomAddF(avgp, s);
}

// edge bias vectors: e is enc_b (first E/2 edges) or enc_W[1]+enc_b (rest);
// fold through pre_W rows [2H:3H] plus pre_b.
__global__ void ebias_kernel(const float* __restrict__ encW, const float* __restrict__ encb,
                             const float* __restrict__ preWc, const float* __restrict__ preb,
                             float* eb0, float* eb1) {
  int f = threadIdx.x;                 // 128
  float a0 = 0.f, a1 = 0.f;
  for (int k = 0; k < Hh; ++k) {
    float w  = preWc[k * Hh + f];
    float e0 = encb[k];
    float e1 = encW[Hh + k] + encb[k]; // enc_W[l][1][k]
    a0 += e0 * w; a1 += e1 * w;
  }
  eb0[f] = a0 + preb[f];
  eb1[f] = a1 + preb[f];
}

// AGG layout per node: [0]=sum, [1]=max, [2]=min, [3]=sumsq, [4]=spare
__global__ void agg_init_kernel(float* __restrict__ AGG) {
  int n = blockIdx.x, f = threadIdx.x;
  float* b = AGG + (size_t)n * (5 * Hh);
  b[0 * Hh + f] = 0.f;
  b[1 * Hh + f] = -INFINITY;
  b[2 * Hh + f] = INFINITY;
  b[3 * Hh + f] = 0.f;
}

__global__ void edge_msg_kernel(const float* __restrict__ Xa, const float* __restrict__ Xb,
                                const int* __restrict__ src, const int* __restrict__ dst,
                                const float* __restrict__ eb0, const float* __restrict__ eb1,
                                float* __restrict__ AGG, int Ehalf) {
  int e = blockIdx.x;
  int f = threadIdx.x;                 // 128
  int s = src[e], d = dst[e];
  float m = Xa[(size_t)d * Hh + f] + Xb[(size_t)s * Hh + f]
          + (e < Ehalf ? eb0[f] : eb1[f]);
  float* base = AGG + (size_t)d * (5 * Hh);
  atomAddF(base + 0 * Hh + f, m);
  atomMaxF(base + 1 * Hh + f, m);
  atomMinF(base + 2 * Hh + f, m);
  atomAddF(base + 3 * Hh + f, m * m);
}

// rewrite AGG in place as [mean, max, min, std, sum]; emit per-node scalers
__global__ void finalize_kernel(float* __restrict__ AGG, const float* __restrict__ deg,
                                const float* __restrict__ avgp,
                                float* __restrict__ amp, float* __restrict__ att) {
  int n = blockIdx.x, f = threadIdx.x;
  float d  = deg[n];
  float dc = fmaxf(d, 1.0f);
  float* base = AGG + (size_t)n * (5 * Hh);
  float s  = base[0 * Hh + f];
  float mx = base[1 * Hh + f];
  float mn = base[2 * Hh + f];
  float s2 = base[3 * Hh + f];
  bool has = d > 0.0f;
  float mean  = s / dc;
  float mean2 = s2 / dc;
  float var = mean2 - mean * mean;
  var = fmaxf(var, 0.0f);
  float sd = sqrtf(var + 1e-5f);
  base[0 * Hh + f] = mean;
  base[1 * Hh + f] = has ? mx : 0.f;
  base[2 * Hh + f] = has ? mn : 0.f;
  base[3 * Hh + f] = sd;
  base[4 * Hh + f] = s;
  if (f == 0) {
    float avg  = avgp[0] / (float)Nn;
    float logd = logf(fmaxf(d, 1.f) + 1.f);
    amp[n] = logd / avg;
    att[n] = avg / logd;
  }
}

// P = T + amp*R2 + att*R3  (per-row scalers folded after the 3 aggregate GEMMs)
__global__ void combine_kernel(const float* __restrict__ T, const float* __restrict__ R2,
                               const float* __restrict__ R3, const float* __restrict__ amp,
                               const float* __restrict__ att, float* __restrict__ P) {
  int n = blockIdx.x, f = threadIdx.x;
  size_t i = (size_t)n * Hh + f;
  P[i] = T[i] + amp[n] * R2[i] + att[n] * R3[i];
}

__global__ void bn_stats_kernel(const float* __restrict__ h, float* bsum, float* bsq) {
  int f = threadIdx.x;                 // 128
  float s = 0.f, q = 0.f;
  for (int n = blockIdx.x; n < Nn; n += gridDim.x) {
    float v = h[(size_t)n * Hh + f];
    s += v; q += v * v;
  }
  atomAddF(&bsum[f], s);
  atomAddF(&bsq[f],  q);
}

__global__ void bn_apply_kernel(float* __restrict__ h, const float* __restrict__ bsum,
                                const float* __restrict__ bsq,
                                const float* __restrict__ g, const float* __restrict__ b) {
  int n = blockIdx.x, f = threadIdx.x;
  size_t i = (size_t)n * Hh + f;
  float mu  = bsum[f] / (float)Nn;
  float var = bsq[f] / (float)Nn - mu * mu;   // biased, matches reference
  float y = g[f] * (h[i] - mu) * rsqrtf(var + 1e-5f) + b[f];
  h[i] = fmaxf(y, 0.f);
}

__global__ void pool_atomic_kernel(const float* __restrict__ h, const int* __restrict__ batch,
                                   float* gsum, float* gmax, float* cnt) {
  int n = blockIdx.x, f = threadIdx.x;
  int g = batch[n];
  float v = h[(size_t)n * Hh + f];
  atomAddF(&gsum[g * Hh + f], v);
  atomMaxF(&gmax[g * Hh + f], v);
  if (f == 0) atomAddF(&cnt[g], 1.f);
}

__global__ void pool_build_kernel(const float* gsum, const float* gmax, const float* cnt,
                                  float* gvec) {
  int g = blockIdx.x, j = threadIdx.x; // 384
  float c = cnt[g];
  float v;
  if (j < Hh)           v = gsum[g * Hh + j] / fmaxf(c, 1.f);
  else if (j < 2 * Hh)  v = (c > 0.f) ? gmax[g * Hh + (j - Hh)] : 0.f;
  else                  v = gsum[g * Hh + (j - 2 * Hh)];
  gvec[g * 3 * Hh + j] = v;
}

// classifier (tiny M=32 GEMMs, scalar)
__global__ void cls1_kernel(const float* gvec, const float* W, const float* b, float* o) {
  int g = blockIdx.x, f = threadIdx.x; // 128
  const float* v = gvec + g * 384;
  float a = b[f];
  for (int k = 0; k < 384; ++k) a += v[k] * W[k * 128 + f];
  o[g * 128 + f] = fmaxf(a, 0.f);
}
__global__ void cls2_kernel(const float* in, const float* W, const float* b, float* o) {
  int g = blockIdx.x, f = threadIdx.x; // 64
  const float* v = in + g * 128;
  float a = b[f];
  for (int k = 0; k < 128; ++k) a += v[k] * W[k * 64 + f];
  o[g * 64 + f] = fmaxf(a, 0.f);
}
__global__ void cls3_kernel(const float* in, const float* W, const float* b, float* o) {
  int g = blockIdx.x, f = threadIdx.x; // 2
  const float* v = in + g * 64;
  float a = b[f];
  for (int k = 0; k < 64; ++k) a += v[k] * W[k * 2 + f];
  o[g * 2 + f] = a;
}

// ---------------- host orchestration ---------------------------------------
extern "C" void kernel_launch(void* const* d_in, const int* in_sizes, int n_in,
                              void* d_out, int out_size, void* d_ws, size_t ws_size,
                              hipStream_t stream) {
  (void)in_sizes; (void)n_in; (void)out_size; (void)ws_size;
  const float* x      = (const float*)d_in[0];
  const float* in_W   = (const float*)d_in[1];
  const float* in_b   = (const float*)d_in[2];
  const float* enc_W  = (const float*)d_in[3];
  const float* enc_b  = (const float*)d_in[4];
  const float* pre_W  = (const float*)d_in[5];
  const float* pre_b  = (const float*)d_in[6];
  const float* post_W = (const float*)d_in[7];
  const float* post_b = (const float*)d_in[8];
  const float* lin_W  = (const float*)d_in[9];
  const float* lin_b  = (const float*)d_in[10];
  const float* bn_g   = (const float*)d_in[11];
  const float* bn_b   = (const float*)d_in[12];
  const float* cW1 = (const float*)d_in[13];
  const float* cb1 = (const float*)d_in[14];
  const float* cW2 = (const float*)d_in[15];
  const float* cb2 = (const float*)d_in[16];
  const float* cW3 = (const float*)d_in[17];
  const float* cb3 = (const float*)d_in[18];
  const int* ei    = (const int*)d_in[19];
  const int* batch = (const int*)d_in[20];
  const int* srcI = ei;
  const int* dstI = ei + Ee;
  float* out = (float*)d_out;

  // workspace bump allocator (~310 MB total; 16B-aligned GEMM buffers first)
  float* w = (float*)d_ws;
  auto alloc = [&](size_t n) { float* p = w; w += n; return p; };
  float* hA   = alloc((size_t)Nn * Hh);
  float* hB   = alloc((size_t)Nn * Hh);
  float* Xa   = alloc((size_t)Nn * Hh);   // also reused as pre-lin buffer P
  float* Xb   = alloc((size_t)Nn * Hh);
  float* Tb   = alloc((size_t)Nn * Hh);
  float* R2   = alloc((size_t)Nn * Hh);
  float* R3   = alloc((size_t)Nn * Hh);
  float* AGG  = alloc((size_t)Nn * 5 * Hh);
  float* deg  = alloc(Nn);
  float* amp  = alloc(Nn);
  float* att  = alloc(Nn);
  float* avgp = alloc(4);
  float* eb0  = alloc(Hh);
  float* eb1  = alloc(Hh);
  float* bnsum = alloc(Hh);
  float* bnsq  = alloc(Hh);
  float* gsum = alloc(Gg * Hh);
  float* gmax = alloc(Gg * Hh);
  float* cnt  = alloc(Gg);
  float* gvec = alloc(Gg * 3 * Hh);
  float* o1   = alloc(Gg * Hh);
  float* o2   = alloc(Gg * (Hh / 2));

  auto fill = [&](float* p, long long n, float v) {
    fill_kernel<<<(int)((n + 255) / 256), 256, 0, stream>>>(p, n, v);
  };

  // degree + avg log-degree statistic
  fill(deg, Nn, 0.f);
  fill(avgp, 1, 0.f);
  deg_kernel<<<(Ee + 255) / 256, 256, 0, stream>>>(dstI, deg, Ee);
  avglog_kernel<<<64, 256, 0, stream>>>(deg, avgp);

  const int gM = (Nn + 127) / 128;
  // h = x @ in_W + in_b
  gemm_wmma_f32<<<gM, 256, 0, stream>>>(x, DIN, in_W, Hh, hA, Hh, Nn, DIN, in_b, 0);

  float* hc = hA; float* hn = hB;
  for (int l = 0; l < Ll; ++l) {
    const float* preW  = pre_W  + (size_t)l * 3  * Hh * Hh;
    const float* postW = post_W + (size_t)l * 16 * Hh * Hh;
    const float* linW  = lin_W  + (size_t)l * Hh * Hh;
    // per-node halves of the pre_nn message
    gemm_wmma_f32<<<gM, 256, 0, stream>>>(hc, Hh, preW,            Hh, Xa, Hh, Nn, Hh, nullptr, 0);
    gemm_wmma_f32<<<gM, 256, 0, stream>>>(hc, Hh, preW + Hh * Hh,  Hh, Xb, Hh, Nn, Hh, nullptr, 0);
    ebias_kernel<<<1, Hh, 0, stream>>>(enc_W + (size_t)l * 2 * Hh, enc_b + (size_t)l * Hh,
                                       preW + 2 * Hh * Hh, pre_b + (size_t)l * Hh, eb0, eb1);
    // edge scatter-aggregate (sum / max / min / sum-of-squares)
    agg_init_kernel<<<Nn, Hh, 0, stream>>>(AGG);
    edge_msg_kernel<<<Ee, Hh, 0, stream>>>(Xa, Xb, srcI, dstI, eb0, eb1, AGG, Ee / 2);
    finalize_kernel<<<Nn, Hh, 0, stream>>>(AGG, deg, avgp, amp, att);
    // post_nn: out = h@Wx + aggr@W1 + amp*(aggr@W2) + att*(aggr@W3) + post_b
    gemm_wmma_f32<<<gM, 256, 0, stream>>>(hc,  Hh,     postW,                        Hh, Tb, Hh, Nn, Hh,     post_b + (size_t)l * Hh, 0);
    gemm_wmma_f32<<<gM, 256, 0, stream>>>(AGG, 5 * Hh, postW + (size_t)1  * Hh * Hh, Hh, Tb, Hh, Nn, 5 * Hh, nullptr, 1);
    gemm_wmma_f32<<<gM, 256, 0, stream>>>(AGG, 5 * Hh, postW + (size_t)6  * Hh * Hh, Hh, R2, Hh, Nn, 5 * Hh, nullptr, 0);
    gemm_wmma_f32<<<gM, 256, 0, stream>>>(AGG, 5 * Hh, postW + (size_t)11 * Hh * Hh, Hh, R3, Hh, Nn, 5 * Hh, nullptr, 0);
    combine_kernel<<<Nn, Hh, 0, stream>>>(Tb, R2, R3, amp, att, Xa);
    // lin
    gemm_wmma_f32<<<gM, 256, 0, stream>>>(Xa, Hh, linW, Hh, hn, Hh, Nn, Hh, lin_b + (size_t)l * Hh, 0);
    // batchnorm (training batch stats) + relu
    fill(bnsum, Hh, 0.f); fill(bnsq, Hh, 0.f);
    bn_stats_kernel<<<256, Hh, 0, stream>>>(hn, bnsum, bnsq);
    bn_apply_kernel<<<Nn, Hh, 0, stream>>>(hn, bnsum, bnsq,
                                           bn_g + (size_t)l * Hh, bn_b + (size_t)l * Hh);
    float* t = hc; hc = hn; hn = t;
  }

  // global mean/max/sum pooling
  fill(gsum, (long long)Gg * Hh, 0.f);
  fill(gmax, (long long)Gg * Hh, -INFINITY);
  fill(cnt, Gg, 0.f);
  pool_atomic_kernel<<<Nn, Hh, 0, stream>>>(hc, batch, gsum, gmax, cnt);
  pool_build_kernel<<<Gg, 3 * Hh, 0, stream>>>(gsum, gmax, cnt, gvec);
  // classifier MLP
  cls1_kernel<<<Gg, Hh, 0, stream>>>(gvec, cW1, cb1, o1);
  cls2_kernel<<<Gg, Hh / 2, 0, stream>>>(o1, cW2, cb2, o2);
  cls3_kernel<<<Gg, 2, 0, stream>>>(o2, cW3, cb3, out);
}